// Encoder_76639396429835
// MI455X (gfx1250) — compile-verified
//
#include <hip/hip_runtime.h>
#include <stdint.h>

#define GLOBAL_AS __attribute__((address_space(1)))
#define LDS_AS    __attribute__((address_space(3)))

typedef __attribute__((ext_vector_type(4))) float v4f;
typedef int v4i __attribute__((vector_size(16)));   // matches builtin param type

constexpr int VOCAB      = 30000;
constexpr int N_PRE      = 20000;
constexpr int VEC_DIM    = 300;
constexpr int INPUT_SIZE = VOCAB - N_PRE;     // 10000
constexpr int HIDDEN     = 256;
constexpr int OUT_DIM    = VEC_DIM + HIDDEN;  // 556
constexpr int WAVES_PER_BLOCK = 8;
constexpr int V4_PER_ROW = VEC_DIM / 4;       // 75

#if defined(__gfx1250__) && \
    __has_builtin(__builtin_amdgcn_global_load_async_to_lds_b128) && \
    __has_builtin(__builtin_amdgcn_global_store_async_from_lds_b128)
#define HAVE_ASYNC_LDS 1
#else
#define HAVE_ASYNC_LDS 0
#endif

__device__ __forceinline__ GLOBAL_AS v4i* as_global_v4(const void* p) {
  // global addrspace(1) pointers are value-identical to generic pointers
  return (GLOBAL_AS v4i*)(uintptr_t)p;
}
__device__ __forceinline__ LDS_AS v4i* as_lds_v4(void* p) {
  // generic -> local addrspacecast (folded by the backend for __shared__ vars)
  return (LDS_AS v4i*)p;
}

__device__ __forceinline__ void wait_async_zero() {
#if defined(__gfx1250__)
#if __has_builtin(__builtin_amdgcn_s_wait_asynccnt)
  __builtin_amdgcn_s_wait_asynccnt(0);
#else
  asm volatile("s_wait_asynccnt 0" ::: "memory");
#endif
#endif
}

extern "C" __global__ void __launch_bounds__(256)
encoder_fused(const int*   __restrict__ batch,
              const float* __restrict__ vectors,
              const float* __restrict__ W,
              const float* __restrict__ bias,
              float*       __restrict__ out,
              int ntok)
{
  __shared__ float smem[WAVES_PER_BLOCK][VEC_DIM + 4];  // 304 floats/wave, 16B-aligned slabs
  const int lane  = threadIdx.x & 31;
  const int wave  = threadIdx.x >> 5;
  const int token = blockIdx.x * WAVES_PER_BLOCK + wave;
  if (token >= ntok) return;

  const int t = batch[token];                 // wave-uniform
  float* orow = out + (size_t)token * OUT_DIM;

  // ---------------- x1: embedding row copy (300 floats) ----------------
  if (t < N_PRE) {
    const float* vrow = vectors + (size_t)t * VEC_DIM;
#if HAVE_ASYNC_LDS
    // DMA vectors row -> LDS (async, 16B per lane), then LDS -> out row.
#pragma unroll
    for (int i = 0; i < 3; ++i) {
      const int idx = lane + 32 * i;
      if (idx < V4_PER_ROW)
        __builtin_amdgcn_global_load_async_to_lds_b128(
            as_global_v4(vrow + 4 * idx), as_lds_v4(&smem[wave][4 * idx]), 0, 0);
    }
    wait_async_zero();  // loads complete => LDS contents valid
#pragma unroll
    for (int i = 0; i < 3; ++i) {
      const int idx = lane + 32 * i;
      if (idx < V4_PER_ROW)
        __builtin_amdgcn_global_store_async_from_lds_b128(
            as_global_v4(orow + 4 * idx), as_lds_v4(&smem[wave][4 * idx]), 0, 0);
    }
    // S_ENDPGM performs an implicit wait-idle; no explicit drain needed.
#else
#pragma unroll
    for (int i = 0; i < 3; ++i) {
      const int idx = lane + 32 * i;
      if (idx < V4_PER_ROW)
        ((v4f*)orow)[idx] = ((const v4f*)vrow)[idx];
    }
#endif
  } else {
    const v4f z = {0.f, 0.f, 0.f, 0.f};
#pragma unroll
    for (int i = 0; i < 3; ++i) {
      const int idx = lane + 32 * i;
      if (idx < V4_PER_ROW) ((v4f*)orow)[idx] = z;
    }
  }

  // ------------- x2: W column gather + bias (256 floats) -------------
  const int h0 = lane * 8;
  const v4f* b4 = (const v4f*)bias;
  v4f lo = b4[lane * 2];
  v4f hi = b4[lane * 2 + 1];
  if (t >= N_PRE) {
    const float* wc = W + (t - N_PRE);   // column c, element (h,c) at wc[h*INPUT_SIZE]
#pragma unroll
    for (int i = 0; i < 4; ++i) lo[i] += wc[(size_t)(h0 + i)     * INPUT_SIZE];
#pragma unroll
    for (int i = 0; i < 4; ++i) hi[i] += wc[(size_t)(h0 + 4 + i) * INPUT_SIZE];
  }
  v4f* o2 = (v4f*)(orow + VEC_DIM);      // byte offset 1200 from 16B-aligned row base
  o2[lane * 2]     = lo;
  o2[lane * 2 + 1] = hi;
}

extern "C" void kernel_launch(void* const* d_in, const int* in_sizes, int n_in,
                              void* d_out, int out_size, void* d_ws, size_t ws_size,
                              hipStream_t stream) {
  (void)n_in; (void)out_size; (void)d_ws; (void)ws_size;
  const int*   batch   = (const int*)  d_in[0];   // [8*512] int32
  const float* vectors = (const float*)d_in[1];   // [20000*300]
  const float* W       = (const float*)d_in[2];   // [256*10000]
  const float* bias    = (const float*)d_in[3];   // [256]
  float*       out     = (float*)d_out;           // [8*512*556]

  const int ntok = in_sizes[0];                   // 4096
  dim3 block(32 * WAVES_PER_BLOCK);               // 256 threads = 8 waves (wave32)
  dim3 grid((ntok + WAVES_PER_BLOCK - 1) / WAVES_PER_BLOCK);
  hipLaunchKernelGGL(encoder_fused, grid, block, 0, stream,
                     batch, vectors, W, bias, out, ntok);
}